// CannyEdgeLayer_37752762532287
// MI455X (gfx1250) — compile-verified
//
#include <hip/hip_runtime.h>
#include <stdint.h>

#define H 512
#define W 512
#define NIMG 16
#define HW (H*W)
#define LOWT  0.1f
#define HIGHT 0.3f

// ---- stage-1 (WMMA blur) tiling: one wave32 per 16x16 output tile ----
#define OT   16           // output tile edge
#define XT1  20           // x tile with halo 2
#define XTS1 21           // padded LDS stride
#define VTS  20           // V tile stride (16 rows x cols 0..19)
#define BT1  18           // blurred tile with halo 1
#define BTS1 19

// ---- hysteresis tiling ----
#define T1  32
#define ET  34
#define ETS 35
#define NSWEEP 32
#define NPASS  16

typedef __attribute__((ext_vector_type(2))) float v2f;
typedef __attribute__((ext_vector_type(8))) float v8f;

// ---- CDNA5 async global->LDS copy (gfx1250), GVS form; ASYNCcnt tracked ----
__device__ __forceinline__ void async_ld_f32(uint32_t lds_byte_off,
                                             const float* base,
                                             uint32_t byte_off) {
  asm volatile("global_load_async_to_lds_b32 %0, %1, %2"
               :: "v"(lds_byte_off), "v"(byte_off), "s"(base)
               : "memory");
}
__device__ __forceinline__ void wait_async_all() {
  asm volatile("s_wait_asynccnt 0" ::: "memory");
}

// Banded [1,2,1]/4 operator: nonzero at k == m+1, m+2, m+3.
__device__ __forceinline__ float band(int k, int m) {
  return (k == m + 2) ? 0.5f : ((k == m + 1 || k == m + 3) ? 0.25f : 0.0f);
}

// -------- Kernel 1: Gaussian blur via chained V_WMMA_F32_16X16X4_F32,
//          then Sobel + magnitude + direction sector (VALU). 1 wave / tile. --
__global__ __launch_bounds__(32)
void canny_stage1(const float* __restrict__ x,
                  float* __restrict__ mag,
                  uint8_t* __restrict__ sect) {
  __shared__ float xt[XT1 * XTS1];   // x tile, halo 2, zero-padded at borders
  __shared__ float vt[OT * VTS];     // vertical-conv intermediate V (16 x 20)
  __shared__ float bt[BT1 * BTS1];   // blurred tile, halo 1

  const int n   = blockIdx.z;
  const int ty0 = blockIdx.y * OT;
  const int tx0 = blockIdx.x * OT;
  const float* xin = x + (size_t)n * HW;
  const int lane = threadIdx.x;      // 0..31 (wave32)
  const int half = lane >> 4, ln = lane & 15;
  const uint32_t lds_x = (uint32_t)(uintptr_t)xt;

  // Async-load x tile rows [ty0-2 .. ty0+17], cols [tx0-2 .. tx0+17];
  // out-of-image entries -> 0 (SAME zero padding of x for the blur).
  for (int e = lane; e < XT1 * XT1; e += 32) {
    const int r = e / XT1, c = e % XT1;
    const int gr = ty0 - 2 + r, gc = tx0 - 2 + c;
    if ((unsigned)gr < (unsigned)H && (unsigned)gc < (unsigned)W) {
      async_ld_f32(lds_x + (uint32_t)(r * XTS1 + c) * 4u, xin,
                   (uint32_t)(gr * W + gc) * 4u);
    } else {
      xt[r * XTS1 + c] = 0.0f;
    }
  }
  wait_async_all();
  __syncthreads();

  // ---- Matmul 1: V(16x16) = Tv(16x20) . X(20x16), K accumulated 5 x 4 ----
  // A layout: lane m = row M; VGPR0/1 hold K = k0+2*half, k0+2*half+1.
  // B layout: row-striped; VGPR0 holds rows k0 / k0+2 in lane halves, VGPR1 k0+1 / k0+3.
  v8f vacc = {};
  for (int q = 0; q < 5; ++q) {
    const int k0 = 4 * q + 2 * half;
    v2f a, b;
    a.x = band(k0,     ln);
    a.y = band(k0 + 1, ln);
    b.x = xt[(k0)     * XTS1 + ln];
    b.y = xt[(k0 + 1) * XTS1 + ln];
    vacc = __builtin_amdgcn_wmma_f32_16x16x4_f32(false, a, false, b,
                                                 (short)0, vacc, false, false);
  }
  // D layout: VGPR v -> (M = v + 8*half, N = ln). Spill V to LDS.
  for (int v = 0; v < 8; ++v)
    vt[(v + 8 * half) * VTS + ln] = vacc[v];
  // V columns 16..18 (outside the 16-wide N range) + zero pad col 19.
  {
    const int cx = 16 + half;        // cols 16,17
    vt[ln * VTS + cx] = 0.25f * xt[(ln + 1) * XTS1 + cx]
                      + 0.5f  * xt[(ln + 2) * XTS1 + cx]
                      + 0.25f * xt[(ln + 3) * XTS1 + cx];
    if (half == 0) {
      vt[ln * VTS + 18] = 0.25f * xt[(ln + 1) * XTS1 + 18]
                        + 0.5f  * xt[(ln + 2) * XTS1 + 18]
                        + 0.25f * xt[(ln + 3) * XTS1 + 18];
      vt[ln * VTS + 19] = 0.0f;      // padded K entry (avoid NaN * 0)
    }
  }
  __syncthreads();

  // ---- Matmul 2: Blur(16x16) = V(16x20) . Th^T(20x16) ----
  v8f bacc = {};
  for (int q = 0; q < 5; ++q) {
    const int k0 = 4 * q + 2 * half;
    v2f a, b;
    a.x = vt[ln * VTS + k0];
    a.y = vt[ln * VTS + k0 + 1];
    b.x = band(k0,     ln);
    b.y = band(k0 + 1, ln);
    bacc = __builtin_amdgcn_wmma_f32_16x16x4_f32(false, a, false, b,
                                                 (short)0, bacc, false, false);
  }
  for (int v = 0; v < 8; ++v)
    bt[(1 + v + 8 * half) * BTS1 + 1 + ln] = bacc[v];

  // Blurred halo ring (68 px) via direct 3x3; zero outside image
  // (the Sobel stage zero-pads *blurred*, matching the reference).
  for (int idx = lane; idx < 68; idx += 32) {
    int rp, cp;
    if (idx < 18)      { rp = -1;       cp = idx - 1;      }
    else if (idx < 36) { rp = 16;       cp = idx - 19;     }
    else if (idx < 52) { rp = idx - 36; cp = -1;           }
    else               { rp = idx - 52; cp = 16;           }
    const int gr = ty0 + rp, gc = tx0 + cp;
    float bv = 0.0f;
    if ((unsigned)gr < (unsigned)H && (unsigned)gc < (unsigned)W) {
      const float* p = &xt[(rp + 1) * XTS1 + (cp + 1)];
      bv = (p[0]           + 2.f * p[1]            + p[2]
          + 2.f * p[XTS1]  + 4.f * p[XTS1 + 1]     + 2.f * p[XTS1 + 2]
          + p[2 * XTS1]    + 2.f * p[2 * XTS1 + 1] + p[2 * XTS1 + 2]) * (1.0f / 16.0f);
    }
    bt[(rp + 1) * BTS1 + (cp + 1)] = bv;
  }
  __syncthreads();

  // ---- Sobel (cross-correlation), magnitude, angle sector: 8 px / lane ----
  for (int k = 0; k < 8; ++k) {
    const int p = lane + 32 * k;          // 0..255 over the 16x16 tile
    const int r = p >> 4, c = p & 15;
    const float* q = &bt[r * BTS1 + c];   // window top-left = (r-1,c-1)
    const float gx = (q[2] - q[0])
                   + 2.f * (q[BTS1 + 2] - q[BTS1])
                   + (q[2 * BTS1 + 2] - q[2 * BTS1]);
    const float gy = (q[2 * BTS1] + 2.f * q[2 * BTS1 + 1] + q[2 * BTS1 + 2])
                   - (q[0]        + 2.f * q[1]            + q[2]);
    const float m = sqrtf(gx * gx + gy * gy);
    float ang = atan2f(gy, gx) * (180.0f / 3.14159f);
    if (ang < 0.0f) ang += 180.0f;
    int s;
    if (ang <= 22.5f || ang > 157.5f)  s = 0;
    else if (ang <= 67.5f)             s = 1;
    else if (ang <= 112.5f)            s = 2;
    else                               s = 3;
    const size_t gi = (size_t)n * HW + (size_t)(ty0 + r) * W + (tx0 + c);
    mag[gi]  = m;
    sect[gi] = (uint8_t)s;
  }
}

// -------- Kernel 2: NMS (wrap-around per jnp.roll) + double threshold ------
__global__ __launch_bounds__(256)
void canny_nms(const float* __restrict__ mag,
               const uint8_t* __restrict__ sect,
               float* __restrict__ edges,
               float* __restrict__ weak) {
  const int idx = blockIdx.x * 256 + threadIdx.x;
  const int n = idx >> 18;                           // HW = 2^18
  const int p = idx & (HW - 1);
  const int i = p >> 9, j = p & (W - 1);
  const float* m = mag + (size_t)n * HW;
  const float v = m[p];
  const int im1 = (i - 1) & (H - 1), ip1 = (i + 1) & (H - 1);
  const int jm1 = (j - 1) & (W - 1), jp1 = (j + 1) & (W - 1);
  float a, b;
  switch (sect[idx]) {
    case 0:  a = m[i * W + jp1];   b = m[i * W + jm1];   break;
    case 1:  a = m[ip1 * W + jm1]; b = m[im1 * W + jp1]; break;
    case 2:  a = m[ip1 * W + j];   b = m[im1 * W + j];   break;
    default: a = m[ip1 * W + jp1]; b = m[im1 * W + jm1]; break;
  }
  const float sup = (v >= a && v >= b) ? v : 0.0f;
  edges[idx] = (sup >= HIGHT) ? 1.0f : 0.0f;
  weak[idx]  = (sup >= LOWT && sup < HIGHT) ? 1.0f : 0.0f;
}

// -------- Kernel 3: one hysteresis pass (in-LDS monotone flood fill) -------
__global__ __launch_bounds__(256)
void canny_hyst(float* __restrict__ edges, const float* __restrict__ weak) {
  __shared__ float et[ET * ETS];
  const int n   = blockIdx.z;
  const int ty0 = blockIdx.y * T1;
  const int tx0 = blockIdx.x * T1;
  float* e        = edges + (size_t)n * HW;
  const float* wk = weak  + (size_t)n * HW;
  const int t = threadIdx.x;
  const uint32_t lds_e = (uint32_t)(uintptr_t)et;

  for (int idx2 = t; idx2 < ET * ET; idx2 += 256) {
    const int r = idx2 / ET, c = idx2 % ET;
    const int gr = (ty0 - 1 + r) & (H - 1);
    const int gc = (tx0 - 1 + c) & (W - 1);
    async_ld_f32(lds_e + (uint32_t)(r * ETS + c) * 4u, e,
                 (uint32_t)(gr * W + gc) * 4u);
  }

  float wreg[4];
  int rr[4], cc[4];
  for (int k = 0; k < 4; ++k) {
    const int p = t + k * 256;
    rr[k] = p >> 5; cc[k] = p & 31;
    wreg[k] = wk[(ty0 + rr[k]) * W + (tx0 + cc[k])];
  }
  wait_async_all();
  __syncthreads();

  for (int it = 0; it < NSWEEP; ++it) {
    for (int k = 0; k < 4; ++k) {
      float* p = &et[(rr[k] + 1) * ETS + (cc[k] + 1)];
      if (wreg[k] > 0.0f && p[0] == 0.0f) {
        const float s = p[-ETS - 1] + p[-ETS] + p[-ETS + 1]
                      + p[-1]                 + p[1]
                      + p[ETS - 1]  + p[ETS]  + p[ETS + 1];
        if (s > 0.0f) p[0] = 1.0f;
      }
    }
    __syncthreads();
  }

  for (int k = 0; k < 4; ++k) {
    e[(ty0 + rr[k]) * W + (tx0 + cc[k])] = et[(rr[k] + 1) * ETS + (cc[k] + 1)];
  }
}

extern "C" void kernel_launch(void* const* d_in, const int* in_sizes, int n_in,
                              void* d_out, int out_size, void* d_ws, size_t ws_size,
                              hipStream_t stream) {
  const float* x = (const float*)d_in[0];
  float* out = (float*)d_out;
  char* ws = (char*)d_ws;
  float*   mag  = (float*)(ws);                                // 16 MB
  float*   weak = (float*)(ws + (size_t)16 * 1024 * 1024);     // 16 MB
  uint8_t* sect = (uint8_t*)(ws + (size_t)32 * 1024 * 1024);   // 4 MB

  const dim3 grid1(W / OT, H / OT, NIMG);     // 32 x 32 x 16, one wave each
  canny_stage1<<<grid1, dim3(32), 0, stream>>>(x, mag, sect);

  canny_nms<<<(NIMG * HW) / 256, dim3(256), 0, stream>>>(mag, sect, out, weak);

  const dim3 grid3(W / T1, H / T1, NIMG);
  for (int p = 0; p < NPASS; ++p) {
    canny_hyst<<<grid3, dim3(256), 0, stream>>>(out, weak);
  }
}